// EnhancedSNN_53609781789168
// MI455X (gfx1250) — compile-verified
//
#include <hip/hip_runtime.h>

// ---------------------------------------------------------------------------
// Hetero-SAGE GNN forward for MI455X (gfx1250, wave32).
//
// Roofline: memory-bound on the layer-2 edge gather/scatter (~4.3 GB -> ~190us
// @ 23.3 TB/s HBM). GEMM total ~60 GFLOP fp32 -> V_WMMA_F32_16X16X4_F32
// (exact fp32 on the matrix pipe). All K-padding is done at LDS staging time
// so the WMMA inner loop has no conditional loads / EXEC churn.
//
// d_in layout (setup_inputs dict order; params flattened in JAX pytree /
// alphabetical-dict order):
//   0..2   x_v, x_e, x_f
//   3..9   ei_vv, ei_ve, ei_vf, ei_ev, ei_ef, ei_fv, ei_fe  ([2,EC]: src then dst)
//   10..12 batch_v, batch_e, batch_f
//   13     num_graphs (==32)
//   14..25 bn: e0b,e0g,e1b,e1g, f0b,f0g,f1b,f1g, v0b,v0g,v1b,v1g
//   26 fc1_b  27 fc1_w  28 fc2_b  29 fc2_w  30 fc_bn_b  31 fc_bn_g
//   32..52 layer0 rels sorted (ef,ev,fe,fv,ve,vf,vv) x (Wl,Wr,b)
//   53..73 layer1 same
// ---------------------------------------------------------------------------

#define HID 128
#define NV_ 100000
#define NE_ 300000
#define NF_ 200000
#define ECNT 600000
#define BGRAPH 32

typedef float v2f __attribute__((ext_vector_type(2)));
typedef float v8f __attribute__((ext_vector_type(8)));

// ---------------- edge scatter: agg[dst] += X[src], cnt[dst] += 1 ----------
__global__ __launch_bounds__(256) void scatter_add_k(
    const float* __restrict__ X, const int* __restrict__ src,
    const int* __restrict__ dst, int nedge, int K,
    float* __restrict__ agg, float* __restrict__ cnt) {
  long long t = (long long)blockIdx.x * blockDim.x + threadIdx.x;
  long long total = (long long)nedge * K;
  if (t >= total) return;
  long long e = t / K;
  int c = (int)(t - e * K);
  int s = src[e], d = dst[e];
  atomicAdd(&agg[(long long)d * K + c], X[(long long)s * K + c]);
  if (c == 0) atomicAdd(&cnt[d], 1.0f);
}

// ---------------------------------------------------------------------------
// C[M,128] += (A[M,K] * rowscale) @ W[K,128], fixed K (multiple of 4).
// 8 wave32s per block; wave w owns cols [16w,16w+16) of a 16-row tile.
// A staged in LDS as interleaved k-pairs: sA[(k/2)*32 + m*2 + (k&1)], so each
// A fragment is a single conflict-free ds_load_b64. B read straight from
// global (W <= 64KB, resident in L2/WGP$). No guards in the MMA loop.
// ---------------------------------------------------------------------------
template <int KF>
__global__ __launch_bounds__(256) void gemm_acc_wmma_fix(
    const float* __restrict__ A, const float* __restrict__ cnt,
    const float* __restrict__ W, float* __restrict__ C, int M) {
  __shared__ float sA[KF * 16];
  __shared__ float sRS[16];
  const int tid = threadIdx.x;
  const int m0 = blockIdx.x * 16;
  if (tid < 16) {
    float rs = 1.0f;
    if (cnt) rs = 1.0f / fmaxf(cnt[m0 + tid], 1.0f);
    sRS[tid] = rs;
  }
  __syncthreads();
  // coalesced over k within each row; scatter into paired LDS layout
  for (int idx = tid; idx < KF * 16; idx += 256) {
    int m = idx / KF, k = idx - m * KF;
    sA[(k >> 1) * 32 + m * 2 + (k & 1)] =
        A[(long long)(m0 + m) * KF + k] * sRS[m];
  }
  __syncthreads();

  const int lane = tid & 31;
  const int n0 = (tid >> 5) * 16;
  const int nn = lane & 15;
  const int khalf = (lane >> 4) << 1;
  const float2* sA2 = reinterpret_cast<const float2*>(sA);
  v8f c = {};
#pragma unroll 8
  for (int k0 = 0; k0 < KF; k0 += 4) {
    const int ka = k0 + khalf;
    float2 av = sA2[(ka >> 1) * 16 + nn];
    v2f a, b;
    a.x = av.x;
    a.y = av.y;
    b.x = W[ka * HID + n0 + nn];
    b.y = W[(ka + 1) * HID + n0 + nn];
    c = __builtin_amdgcn_wmma_f32_16x16x4_f32(false, a, false, b,
                                              (short)0, c, false, false);
  }
  const int rbase = m0 + ((lane >> 4) << 3);
  const int col = n0 + nn;
#pragma unroll
  for (int i = 0; i < 8; ++i) {
    long long off = (long long)(rbase + i) * HID + col;
    C[off] += c[i];
  }
}

// ---------------------------------------------------------------------------
// Small-K variant (K <= 8, runtime). Both A and W tiles are zero-padded to
// Kpad (4 or 8) in LDS during staging; MMA loop is guard-free.
// ---------------------------------------------------------------------------
__global__ __launch_bounds__(256) void gemm_acc_wmma_small(
    const float* __restrict__ A, const float* __restrict__ cnt,
    const float* __restrict__ W, float* __restrict__ C, int M, int K) {
  __shared__ float sA[8 * 16];
  __shared__ float sW[8 * HID];
  __shared__ float sRS[16];
  const int tid = threadIdx.x;
  const int m0 = blockIdx.x * 16;
  const int Kpad = (K + 3) & ~3;
  if (tid < 16) {
    float rs = 1.0f;
    if (cnt) rs = 1.0f / fmaxf(cnt[m0 + tid], 1.0f);
    sRS[tid] = rs;
  }
  __syncthreads();
  for (int idx = tid; idx < Kpad * 16; idx += 256) {
    int m = idx / Kpad, k = idx - m * Kpad;
    float v = 0.0f;
    if (k < K) v = A[(long long)(m0 + m) * K + k] * sRS[m];
    sA[(k >> 1) * 32 + m * 2 + (k & 1)] = v;
  }
  for (int idx = tid; idx < Kpad * HID; idx += 256) {
    int k = idx >> 7;
    sW[idx] = (k < K) ? W[idx] : 0.0f;
  }
  __syncthreads();

  const int lane = tid & 31;
  const int n0 = (tid >> 5) * 16;
  const int nn = lane & 15;
  const int khalf = (lane >> 4) << 1;
  const float2* sA2 = reinterpret_cast<const float2*>(sA);
  v8f c = {};
  for (int k0 = 0; k0 < Kpad; k0 += 4) {
    const int ka = k0 + khalf;
    float2 av = sA2[(ka >> 1) * 16 + nn];
    v2f a, b;
    a.x = av.x;
    a.y = av.y;
    b.x = sW[ka * HID + n0 + nn];
    b.y = sW[(ka + 1) * HID + n0 + nn];
    c = __builtin_amdgcn_wmma_f32_16x16x4_f32(false, a, false, b,
                                              (short)0, c, false, false);
  }
  const int rbase = m0 + ((lane >> 4) << 3);
  const int col = n0 + nn;
#pragma unroll
  for (int i = 0; i < 8; ++i) {
    long long off = (long long)(rbase + i) * HID + col;
    C[off] += c[i];
  }
}

// ---------------- out[i] = a[i] (+ b[i]) (+ c[i]) --------------------------
__global__ __launch_bounds__(256) void sum3_k(
    const float* __restrict__ a, const float* __restrict__ b,
    const float* __restrict__ c, float* __restrict__ out, int n) {
  int i = blockIdx.x * blockDim.x + threadIdx.x;
  if (i >= n) return;
  float v = a[i];
  if (b) v += b[i];
  if (c) v += c[i];
  out[i] = v;
}

// --------- acc = (acc + bsum[c]) * invk (+ resid) --------------------------
__global__ __launch_bounds__(256) void combine_k(
    float* __restrict__ acc, const float* __restrict__ bsum,
    const float* __restrict__ resid, float invk, long long total) {
  long long i = (long long)blockIdx.x * blockDim.x + threadIdx.x;
  if (i >= total) return;
  int ch = (int)(i & (HID - 1));
  float v = (acc[i] + bsum[ch]) * invk;
  if (resid) v += resid[i];
  acc[i] = v;
}

// ---------------- per-channel sum / sumsq over n rows ----------------------
__global__ __launch_bounds__(HID) void bn_stats_k(
    const float* __restrict__ X, long long n, float* __restrict__ sums) {
  const int c = threadIdx.x;
  float s = 0.f, s2 = 0.f;
  for (long long r = blockIdx.x; r < n; r += gridDim.x) {
    float v = X[r * HID + c];
    s += v;
    s2 += v * v;
  }
  atomicAdd(&sums[c], s);
  atomicAdd(&sums[HID + c], s2);
}

__global__ __launch_bounds__(HID) void bn_finalize_k(
    const float* __restrict__ sums, const float* __restrict__ g,
    const float* __restrict__ b, float invn, float* __restrict__ ss) {
  const int c = threadIdx.x;
  float mean = sums[c] * invn;
  float var = sums[HID + c] * invn - mean * mean;
  float sc = g[c] * rsqrtf(var + 1e-5f);
  ss[c] = sc;
  ss[HID + c] = b[c] - mean * sc;
}

__global__ __launch_bounds__(256) void bn_apply_relu_k(
    float* __restrict__ X, const float* __restrict__ ss, long long total) {
  long long i = (long long)blockIdx.x * blockDim.x + threadIdx.x;
  if (i >= total) return;
  int c = (int)(i & (HID - 1));
  float v = X[i] * ss[c] + ss[HID + c];
  X[i] = fmaxf(v, 0.0f);
}

// ---------------- graph pooling (segment mean) -----------------------------
__global__ __launch_bounds__(256) void pool_accum_k(
    const float* __restrict__ X, const int* __restrict__ batch, long long n,
    float* __restrict__ psum, float* __restrict__ pcnt, int colOff) {
  long long i = (long long)blockIdx.x * blockDim.x + threadIdx.x;
  if (i >= n * HID) return;
  long long node = i >> 7;
  int c = (int)(i & (HID - 1));
  int b = batch[node];
  atomicAdd(&psum[(long long)b * (3 * HID) + colOff + c], X[i]);
  if (c == 0) atomicAdd(&pcnt[b], 1.0f);
}

__global__ __launch_bounds__(256) void pool_final_k(
    const float* __restrict__ psum, const float* __restrict__ pcnt,
    float* __restrict__ pooled) {
  int i = blockIdx.x * blockDim.x + threadIdx.x;
  if (i >= BGRAPH * 3 * HID) return;
  int b = i / (3 * HID);
  int col = i - b * (3 * HID);
  int t = col >> 7;
  pooled[i] = psum[i] / fmaxf(pcnt[t * BGRAPH + b], 1.0f);
}

// ---------------- FC head --------------------------------------------------
__global__ __launch_bounds__(256) void fc1_k(
    const float* __restrict__ pooled, const float* __restrict__ w,
    const float* __restrict__ bias, float* __restrict__ out) {
  int idx = blockIdx.x * blockDim.x + threadIdx.x;  // 32*256
  int r = idx >> 8, c = idx & 255;
  float s = bias[c];
  for (int k = 0; k < 3 * HID; ++k) s += pooled[r * (3 * HID) + k] * w[k * 256 + c];
  out[idx] = s;
}

__global__ __launch_bounds__(256) void fc_bn_relu_k(
    float* __restrict__ h, const float* __restrict__ g,
    const float* __restrict__ b) {
  int c = threadIdx.x;  // 256 channels
  float s = 0.f, s2 = 0.f;
  for (int r = 0; r < BGRAPH; ++r) {
    float v = h[r * 256 + c];
    s += v;
    s2 += v * v;
  }
  float m = s / (float)BGRAPH;
  float var = s2 / (float)BGRAPH - m * m;
  float sc = g[c] * rsqrtf(var + 1e-5f);
  float sh = b[c] - m * sc;
  for (int r = 0; r < BGRAPH; ++r) {
    float v = h[r * 256 + c] * sc + sh;
    h[r * 256 + c] = fmaxf(v, 0.0f);
  }
}

__global__ __launch_bounds__(64) void fc2_k(
    const float* __restrict__ h, const float* __restrict__ w,
    const float* __restrict__ bias, float* __restrict__ out) {
  int idx = blockIdx.x * blockDim.x + threadIdx.x;  // 32*10
  if (idx >= BGRAPH * 10) return;
  int r = idx / 10, c = idx - r * 10;
  float s = bias[c];
  for (int k = 0; k < 256; ++k) s += h[r * 256 + k] * w[k * 10 + c];
  out[idx] = s;
}

// ---------------------------------------------------------------------------
extern "C" void kernel_launch(void* const* d_in, const int* in_sizes, int n_in,
                              void* d_out, int out_size, void* d_ws, size_t ws_size,
                              hipStream_t stream) {
  (void)in_sizes; (void)n_in; (void)out_size; (void)ws_size;

  const long long nsz[3] = {NV_, NE_, NF_};
  const int krel[3] = {3, 2, 2};                         // #relations into v,e,f
  const long long nodeoff[3] = {0, NV_, NV_ + NE_};
  const int bnbase[3] = {22, 14, 18};                    // v,e,f: leaf idx of L0 'b'
  struct RelT { int ei, s, d, pb; };
  const RelT rels[7] = {
      {3, 0, 0, 50}, {6, 1, 0, 35}, {8, 2, 0, 41},       // dst v: vv, ev, fv
      {4, 0, 1, 44}, {9, 2, 1, 38},                      // dst e: ve, fe
      {5, 0, 2, 47}, {7, 1, 2, 32},                      // dst f: vf, ef
  };

  const long long NTOT = NV_ + NE_ + NF_;
  float* hA     = (float*)d_ws;                          // 600k x 128
  float* hB     = hA + NTOT * HID;                       // 600k x 128
  float* agg    = hB + NTOT * HID;                       // 300k x 128 scratch
  float* cnt    = agg + (long long)NE_ * HID;            // 300k
  float* wsum   = cnt + NE_;                             // 128 x 128
  float* bsum   = wsum + HID * HID;                      // 128
  float* bnsums = bsum + HID;                            // 256
  float* ssbuf  = bnsums + 2 * HID;                      // 256
  float* psum   = ssbuf + 2 * HID;                       // 32 x 384
  float* pcnt   = psum + BGRAPH * 3 * HID;               // 96
  float* pooled = pcnt + 3 * BGRAPH;                     // 32 x 384
  float* fctmp  = pooled + BGRAPH * 3 * HID;             // 32 x 256

  for (int L = 0; L < 2; ++L) {
    const float* xin[3];
    int kin[3];
    float* accb;
    const float* residb;
    if (L == 0) {
      xin[0] = (const float*)d_in[0]; xin[1] = (const float*)d_in[1];
      xin[2] = (const float*)d_in[2];
      kin[0] = 7; kin[1] = 2; kin[2] = 5;
      accb = hA; residb = nullptr;
    } else {
      for (int t = 0; t < 3; ++t) { xin[t] = hA + nodeoff[t] * HID; kin[t] = HID; }
      accb = hB; residb = hA;
    }

    for (int t = 0; t < 3; ++t) {
      const long long n = nsz[t];
      float* acc = accb + nodeoff[t] * HID;
      hipMemsetAsync(acc, 0, (size_t)(n * HID) * sizeof(float), stream);

      const float* WrP[3] = {nullptr, nullptr, nullptr};
      const float* bP[3]  = {nullptr, nullptr, nullptr};
      int nr = 0;
      for (int r = 0; r < 7; ++r) {
        if (rels[r].d != t) continue;
        const int ks = kin[rels[r].s];
        const int* ei = (const int*)d_in[rels[r].ei];
        const int pb = rels[r].pb + 21 * L;
        hipMemsetAsync(agg, 0, (size_t)(n * ks) * sizeof(float), stream);
        hipMemsetAsync(cnt, 0, (size_t)n * sizeof(float), stream);
        const long long tot = (long long)ECNT * ks;
        scatter_add_k<<<(unsigned)((tot + 255) / 256), 256, 0, stream>>>(
            xin[rels[r].s], ei, ei + ECNT, ECNT, ks, agg, cnt);
        if (ks == HID) {
          gemm_acc_wmma_fix<HID><<<(unsigned)(n / 16), 256, 0, stream>>>(
              agg, cnt, (const float*)d_in[pb], acc, (int)n);
        } else {
          gemm_acc_wmma_small<<<(unsigned)(n / 16), 256, 0, stream>>>(
              agg, cnt, (const float*)d_in[pb], acc, (int)n, ks);
        }
        WrP[nr] = (const float*)d_in[pb + 1];
        bP[nr]  = (const float*)d_in[pb + 2];
        ++nr;
      }
      // combined self term: x_t @ (sum_r Wr_r) + sum_r b_r
      sum3_k<<<(kin[t] * HID + 255) / 256, 256, 0, stream>>>(
          WrP[0], WrP[1], WrP[2], wsum, kin[t] * HID);
      sum3_k<<<1, HID, 0, stream>>>(bP[0], bP[1], bP[2], bsum, HID);
      if (kin[t] == HID) {
        gemm_acc_wmma_fix<HID><<<(unsigned)(n / 16), 256, 0, stream>>>(
            xin[t], nullptr, wsum, acc, (int)n);
      } else {
        gemm_acc_wmma_small<<<(unsigned)(n / 16), 256, 0, stream>>>(
            xin[t], nullptr, wsum, acc, (int)n, kin[t]);
      }

      const float* res = residb ? residb + nodeoff[t] * HID : nullptr;
      const long long total = n * HID;
      combine_k<<<(unsigned)((total + 255) / 256), 256, 0, stream>>>(
          acc, bsum, res, 1.0f / (float)krel[t], total);

      // BatchNorm (population stats over all n rows) + ReLU, in place
      hipMemsetAsync(bnsums, 0, 2 * HID * sizeof(float), stream);
      bn_stats_k<<<512, HID, 0, stream>>>(acc, n, bnsums);
      bn_finalize_k<<<1, HID, 0, stream>>>(
          bnsums, (const float*)d_in[bnbase[t] + 2 * L + 1],
          (const float*)d_in[bnbase[t] + 2 * L], 1.0f / (float)n, ssbuf);
      bn_apply_relu_k<<<(unsigned)((total + 255) / 256), 256, 0, stream>>>(
          acc, ssbuf, total);
    }
  }

  // segment-mean pooling per graph, concat [B, 3*HID]
  hipMemsetAsync(psum, 0, BGRAPH * 3 * HID * sizeof(float), stream);
  hipMemsetAsync(pcnt, 0, 3 * BGRAPH * sizeof(float), stream);
  for (int t = 0; t < 3; ++t) {
    const long long tot = nsz[t] * HID;
    pool_accum_k<<<(unsigned)((tot + 255) / 256), 256, 0, stream>>>(
        hB + nodeoff[t] * HID, (const int*)d_in[10 + t], nsz[t], psum,
        pcnt + t * BGRAPH, t * HID);
  }
  pool_final_k<<<(BGRAPH * 3 * HID + 255) / 256, 256, 0, stream>>>(psum, pcnt, pooled);

  // FC head: fc1 -> BN -> ReLU -> fc2
  fc1_k<<<BGRAPH, 256, 0, stream>>>(pooled, (const float*)d_in[27],
                                    (const float*)d_in[26], fctmp);
  fc_bn_relu_k<<<1, 256, 0, stream>>>(fctmp, (const float*)d_in[31],
                                      (const float*)d_in[30]);
  fc2_k<<<(BGRAPH * 10 + 63) / 64, 64, 0, stream>>>(
      fctmp, (const float*)d_in[29], (const float*)d_in[28], (float*)d_out);
}